// phys_inform_net_7567732375987
// MI455X (gfx1250) — compile-verified
//
#include <hip/hip_runtime.h>
#include <math.h>

// FNO forward for MI455X (gfx1250, wave32).
// GEMM-shaped contractions (spectral partial-DFTs, 1x1 conv, fc1) run on
// V_WMMA_F32_16X16X4_F32 (fp32 matrix op). Each wave computes NT adjacent
// 16x16 C tiles sharing a single A fetch per K-step (cuts streamed-operand
// traffic and raises WMMA density). Small mode-space contractions are VALU
// kernels; fc1+gelu+fc2+residual is fused per 16-pixel wave tile.

#define NP     265        // padded spatial (256 + 9)
#define NP2    70225      // 265*265
#define MODES  12
#define KM     24         // retained k rows (0..11, 253..264)
#define WIDTH  32
#define BATCHN 8
#define HH     256

typedef float v2f __attribute__((ext_vector_type(2)));
typedef float v8f __attribute__((ext_vector_type(8)));

__device__ __forceinline__ v8f wmma_f32_16x16x4(v2f a, v2f b, v8f c) {
#if __has_builtin(__builtin_amdgcn_wmma_f32_16x16x4_f32)
  // (neg_a, A, neg_b, B, c_mod, C, reuse_a, reuse_b)
  return __builtin_amdgcn_wmma_f32_16x16x4_f32(false, a, false, b, (short)0, c,
                                               false, false);
#else
  for (int j = 0; j < 8; ++j) c[j] += a.x * b.x + a.y * b.y;  // hedge only
  return c;
#endif
}

__device__ __forceinline__ float gelu_exact(float x) {
  return 0.5f * x * (1.0f + erff(x * 0.7071067811865475f));
}

// ---------------------------------------------------------------------------
// Twiddle tables (phase reduced exactly: angle = 2*pi*((k*h) mod 265)/265).
// E  : [268][32]  forward w-DFT (cols 0..11 cos, 12..23 -sin, rest zero pad)
// Fk : [24][265][2]  forward h-DFT (cos, sin), k in {0..11, 253..264}
// G  : [265][48]  inverse h-DFT (col 2k = cos, col 2k+1 = -sin)
// ---------------------------------------------------------------------------
__global__ void k_tables(float* __restrict__ E, float* __restrict__ Fk,
                         float* __restrict__ G) {
  int i = blockIdx.x * blockDim.x + threadIdx.x;
  const float step = 6.28318530717958647692f / 265.0f;
  if (i < 268 * 32) {
    int m = i & 31, w = i >> 5;
    float v = 0.0f;
    if (w < NP) {
      if (m < MODES) {
        int p = (m * w) % NP; v = cosf(step * (float)p);
      } else if (m < 2 * MODES) {
        int p = ((m - MODES) * w) % NP; v = -sinf(step * (float)p);
      }
    }
    E[i] = v;
  }
  if (i < KM * NP) {
    int h = i % NP, kidx = i / NP;
    int kv = (kidx < MODES) ? kidx : (NP - KM + kidx);  // 253..264
    int p = (kv * h) % NP;
    Fk[2 * i]     = cosf(step * (float)p);
    Fk[2 * i + 1] = sinf(step * (float)p);
  }
  if (i < NP * 2 * KM) {
    int c = i % (2 * KM), h = i / (2 * KM);
    int kidx = c >> 1;
    int kv = (kidx < MODES) ? kidx : (NP - KM + kidx);
    int p = (kv * h) % NP;
    G[i] = (c & 1) ? -sinf(step * (float)p) : cosf(step * (float)p);
  }
}

// fc0 channel lift (3->32) + zero pad to 265x265, channels-first layout.
__global__ void k_fc0_pad(const float* __restrict__ x_en,
                          const float* __restrict__ w,
                          const float* __restrict__ bias,
                          float* __restrict__ out, int total) {
  int i = blockIdx.x * blockDim.x + threadIdx.x;
  if (i >= total) return;
  int wq = i % NP;
  int t = i / NP;
  int hq = t % NP; t /= NP;
  int d = t % WIDTH;
  int b = t / WIDTH;
  float v = 0.0f;
  if (hq < HH && wq < HH) {
    long base = (((long)b * 3) << 16) + (hq << 8) + wq;
    v = bias[d] + x_en[base]          * w[d]
                + x_en[base + 65536]  * w[32 + d]
                + x_en[base + 131072] * w[64 + d];
  }
  out[i] = v;
}

// ---------------------------------------------------------------------------
// fp32 WMMA GEMM, wave computes a 16 x (16*NT) C strip (one A fetch feeds NT
// WMMAs per K-step):  C = alpha*(A x B) [+D][+biasRow][+biasCol][gelu]
// A: MxK row-major (lda), B: KxN row-major (ldb), C/D: MxN (ldc). K % 4 == 0;
// A rows may be over-read by <=3 elements (caller guarantees allocation).
// M/N edges: loads clamped, stores masked (EXEC stays all-ones for WMMA).
// ---------------------------------------------------------------------------
template <int NT>
__global__ void k_gemm(const float* __restrict__ A, const float* __restrict__ B,
                       float* __restrict__ C, const float* __restrict__ D,
                       const float* __restrict__ biasRow,
                       const float* __restrict__ biasCol,
                       int M, int N, int K, int lda, int ldb, int ldc,
                       long sA, long sB, long sC,
                       int batch, float alpha, int doGelu) {
  int wid  = blockIdx.x * (blockDim.x >> 5) + (threadIdx.x >> 5);
  int lane = threadIdx.x & 31;
  int tn  = (N + 15) >> 4;                 // 16-wide column tiles
  int tgn = (tn + NT - 1) / NT;            // groups of NT tiles
  int tilesPer = ((M + 15) >> 4) * tgn;
  int bb = wid / tilesPer;
  if (bb >= batch) return;                 // wave-uniform exit
  int t = wid - bb * tilesPer;
  int row0 = (t / tgn) << 4;
  int col0 = (t % tgn) * (NT << 4);
  const float* Ab = A + sA * bb;
  const float* Bb = B + sB * bb;
  float*       Cb = C + sC * bb;
  int half = lane >> 4, lm = lane & 15;
  int ar = row0 + lm; if (ar > M - 1) ar = M - 1;
  // A 16x4: lane = M row, VGPR pair covers K = k0+2*half, k0+2*half+1
  const float* Ap = Ab + (long)ar * lda + (half << 1);
  // B 4x16 per sub-tile: lane = N col, VGPR j holds K row (k0 + 2*half + j)
  const float* Bp[NT];
#pragma unroll
  for (int s = 0; s < NT; ++s) {
    int bc = col0 + (s << 4) + lm; if (bc > N - 1) bc = N - 1;
    Bp[s] = Bb + (long)(half << 1) * ldb + bc;
  }
  v8f acc[NT];
#pragma unroll
  for (int s = 0; s < NT; ++s) acc[s] = (v8f){0.f,0.f,0.f,0.f,0.f,0.f,0.f,0.f};
  for (int k0 = 0; k0 < K; k0 += 4) {
    v2f a;
    a.x = Ap[0]; a.y = Ap[1];
    __builtin_prefetch(Ap + 16, 0, 1);     // global_prefetch on the A stream
#pragma unroll
    for (int s = 0; s < NT; ++s) {
      v2f b;
      b.x = Bp[s][0]; b.y = Bp[s][ldb];
      acc[s] = wmma_f32_16x16x4(a, b, acc[s]);
      Bp[s] += (long)4 * ldb;
    }
    Ap += 4;
  }
  // C 16x16: VGPR j -> row j + 8*half, col = lane&15
#pragma unroll
  for (int s = 0; s < NT; ++s) {
    int cc = col0 + (s << 4) + lm;
    if (cc >= N) continue;
    const float* Db = D ? (D + sC * bb) : (const float*)0;
    for (int j = 0; j < 8; ++j) {
      int cr = row0 + j + (half << 3);
      if (cr < M) {
        float v = acc[s][j] * alpha;
        if (Db)      v += Db[(long)cr * ldc + cc];
        if (biasRow) v += biasRow[cr];
        if (biasCol) v += biasCol[cc];
        if (doGelu)  v = gelu_exact(v);
        Cb[(long)cr * ldc + cc] = v;
      }
    }
  }
}

// Stage-2 forward DFT along h: Xft[bi,k,m] = sum_h A1[bi,h,{re,im}m] * e^{-i2pi k h/N}
__global__ void k_kdft(const float* __restrict__ A1, const float* __restrict__ Fk,
                       float* __restrict__ Xft) {
  int i = blockIdx.x * blockDim.x + threadIdx.x;
  if (i >= BATCHN * WIDTH * KM * MODES) return;
  int m = i % MODES;
  int t = i / MODES;
  int kidx = t % KM;
  int bi = t / KM;
  const float* ar = A1 + (long)bi * NP * 32 + m;   // re at +0, im at +MODES
  const float* fk = Fk + (long)kidx * NP * 2;
  float re = 0.f, im = 0.f;
  for (int h = 0; h < NP; ++h) {
    float xr = ar[0], xi = ar[MODES];
    float c = fk[0], s = fk[1];
    re += xr * c + xi * s;
    im += xi * c - xr * s;
    ar += 32; fk += 2;
  }
  Xft[2 * (long)i]     = re;
  Xft[2 * (long)i + 1] = im;
}

// Complex channel mix: Oft[b,o,k,m] = sum_i Xft[b,i,k,m] * W[i,o,kx,m]
__global__ void k_mix(const float* __restrict__ Xft,
                      const float* __restrict__ w1r, const float* __restrict__ w1i,
                      const float* __restrict__ w2r, const float* __restrict__ w2i,
                      float* __restrict__ Oft, int layer) {
  int i = blockIdx.x * blockDim.x + threadIdx.x;
  if (i >= BATCHN * WIDTH * KM * MODES) return;
  int m = i % MODES;
  int t = i / MODES;
  int kidx = t % KM;
  int o = (t / KM) % WIDTH;
  int b = t / (KM * WIDTH);
  const float *wr, *wi;
  int kx;
  if (kidx < MODES) { kx = kidx;         wr = w1r; wi = w1i; }
  else              { kx = kidx - MODES; wr = w2r; wi = w2i; }
  // weight (L,32,32,12,12): (((l*32+i)*32+o)*12+kx)*12+m ; i-stride 4608
  long wb = ((long)layer * 1024 + o) * 144 + kx * 12 + m;
  const float* xf = Xft + (((long)b * WIDTH) * KM + kidx) * (MODES * 2) + 2 * m;
  float orr = 0.f, oii = 0.f;
  for (int ic = 0; ic < WIDTH; ++ic) {
    float xr = xf[0], xi = xf[1];
    float a = wr[wb], bb = wi[wb];
    orr += xr * a - xi * bb;
    oii += xr * bb + xi * a;
    xf += (long)KM * MODES * 2;
    wb += 4608;
  }
  Oft[2 * (long)i]     = orr;
  Oft[2 * (long)i + 1] = oii;
}

// Inverse along w: Bw[bo, 2k(+1), w] = sum_m c_m * Oft[bo,k,m] * e^{+i2pi m w/N}
// (c_0 = 1, c_m = 2: conjugate-symmetric half of irfft)
__global__ void k_invw(const float* __restrict__ Oft, float* __restrict__ Bw) {
  int i = blockIdx.x * blockDim.x + threadIdx.x;
  if (i >= BATCHN * WIDTH * KM * NP) return;
  int w = i % NP;
  int t = i / NP;
  int kidx = t % KM;
  int bo = t / KM;
  const float* of = Oft + ((long)bo * KM + kidx) * (MODES * 2);
  const float step = 6.28318530717958647692f / 265.0f;
  float re = 0.f, im = 0.f;
  for (int m = 0; m < MODES; ++m) {
    float cm = (m == 0) ? 1.0f : 2.0f;
    int p = (m * w) % NP;
    float s, c;
    sincosf(step * (float)p, &s, &c);
    float orr = of[2 * m], oii = of[2 * m + 1];
    re += cm * (orr * c - oii * s);
    im += cm * (oii * c + orr * s);
  }
  long ob = ((long)bo * (2 * KM) + 2 * kidx) * NP + w;
  Bw[ob]      = re;
  Bw[ob + NP] = im;
}

// Fused head: per 16-pixel wave tile: h = gelu(x*fc1 + b1) via WMMA (16x128,K=32),
// then fc2 (128->3) + residual, writing pred directly.
__global__ void __launch_bounds__(128)
k_head(const float* __restrict__ X, const float* __restrict__ fc1w,
       const float* __restrict__ fc1b, const float* __restrict__ fc2w,
       const float* __restrict__ fc2b, const float* __restrict__ x_en,
       float* __restrict__ pred) {
  __shared__ float hsm[4][16 * 128];
  int wv = threadIdx.x >> 5, lane = threadIdx.x & 31;
  int tile = blockIdx.x * 4 + wv;
  int p0 = tile << 4;
  int half = lane >> 4, lm = lane & 15;
  int p = p0 + lm;
  int b = p >> 16;
  int hw = p & 65535;
  int hh = hw >> 8, ww = hw & 255;
  // lane's pixel row of A (16x32 tile), channel stride NP2, K pair via half
  const float* Abase =
      X + (((long)b * WIDTH) * NP + hh) * NP + ww + (long)(half << 1) * NP2;
  for (int nt = 0; nt < 8; ++nt) {
    v8f acc = {0.f, 0.f, 0.f, 0.f, 0.f, 0.f, 0.f, 0.f};
    const float* Ap = Abase;
    const float* Bp = fc1w + (long)(half << 1) * 128 + nt * 16 + lm;
    for (int k0 = 0; k0 < 32; k0 += 4) {
      v2f a, bv;
      a.x  = Ap[0];  a.y  = Ap[NP2];
      bv.x = Bp[0];  bv.y = Bp[128];
      acc = wmma_f32_16x16x4(a, bv, acc);
      Ap += (long)4 * NP2;
      Bp += 4 * 128;
    }
    int cc = nt * 16 + lm;
    for (int j = 0; j < 8; ++j) {
      int cr = j + (half << 3);
      hsm[wv][cr * 128 + cc] = gelu_exact(acc[j] + fc1b[cc]);
    }
  }
  __syncthreads();
  for (int t = lane; t < 48; t += 32) {
    int pi = t & 15, oc = t >> 4;
    const float* hr = &hsm[wv][pi * 128];
    float s = fc2b[oc];
#pragma unroll 8
    for (int k = 0; k < 128; ++k) s += hr[k] * fc2w[k * 3 + oc];
    int pp = p0 + pi;
    long oidx = (((long)(pp >> 16) * 3 + oc) << 16) + (pp & 65535);
    pred[oidx] = s + x_en[oidx];
  }
}

// pred_de / x_de strided downsamples + x_en passthrough.
__global__ void k_finalize(const float* __restrict__ x_en,
                           const float* __restrict__ pred,
                           float* __restrict__ pred_de, float* __restrict__ x_de,
                           float* __restrict__ x_en_out) {
  int i = blockIdx.x * blockDim.x + threadIdx.x;
  const int nDe = BATCHN * 3 * 64 * 64;
  if (i < nDe) {
    int w4 = i & 63;
    int t = i >> 6;
    int h4 = t & 63;
    t >>= 6;  // t = b*3 + c
    long src = ((long)t << 16) + ((h4 * 4) << 8) + (w4 * 4);
    pred_de[i] = pred[src];
    x_de[i]    = x_en[src];
  }
  if (i < BATCHN * 3 * 65536) x_en_out[i] = x_en[i];
}

// ---------------------------------------------------------------------------
extern "C" void kernel_launch(void* const* d_in, const int* in_sizes, int n_in,
                              void* d_out, int out_size, void* d_ws,
                              size_t ws_size, hipStream_t stream) {
  (void)in_sizes; (void)n_in; (void)out_size; (void)ws_size;
  const float* x_en  = (const float*)d_in[0];
  const float* fc0_w = (const float*)d_in[2];
  const float* fc0_b = (const float*)d_in[3];
  const float* w1r   = (const float*)d_in[4];
  const float* w1i   = (const float*)d_in[5];
  const float* w2r   = (const float*)d_in[6];
  const float* w2i   = (const float*)d_in[7];
  const float* pw_w  = (const float*)d_in[8];
  const float* pw_b  = (const float*)d_in[9];
  const float* fc1_w = (const float*)d_in[10];
  const float* fc1_b = (const float*)d_in[11];
  const float* fc2_w = (const float*)d_in[12];
  const float* fc2_b = (const float*)d_in[13];
  float* out = (float*)d_out;

  // workspace layout (floats); ~167 MB total
  float* ws = (float*)d_ws;
  const long SZ_X = (long)BATCHN * WIDTH * NP2;           // 17,977,600
  float* xA  = ws; ws += SZ_X;
  float* xS  = ws; ws += SZ_X;
  float* A1  = ws; ws += (long)BATCHN * WIDTH * NP * 32;  // stage-1 DFT out
  float* Bw  = ws; ws += (long)BATCHN * WIDTH * 2 * KM * NP;
  float* Xft = ws; ws += (long)BATCHN * WIDTH * KM * MODES * 2;
  float* Oft = ws; ws += (long)BATCHN * WIDTH * KM * MODES * 2;
  float* Et  = ws; ws += 268 * 32;
  float* Fk  = ws; ws += KM * NP * 2;
  float* Gt  = ws; ws += NP * 2 * KM;

  k_tables<<<(12720 + 255) / 256, 256, 0, stream>>>(Et, Fk, Gt);
  k_fc0_pad<<<(int)((SZ_X + 255) / 256), 256, 0, stream>>>(x_en, fc0_w, fc0_b,
                                                           xA, (int)SZ_X);

  float* cur = xA;
  float* nxt = xS;
  for (int l = 0; l < 4; ++l) {
    {  // stage-1 w-DFT GEMM: [67840 x 268] x [268 x 32] -> A1 (NT=2: one pass)
      int M = BATCHN * WIDTH * NP, N = 32, K = 268;
      long waves = (long)((M + 15) / 16) * (((N + 15) / 16 + 1) / 2);
      k_gemm<2><<<(int)((waves * 32 + 255) / 256), 256, 0, stream>>>(
          cur, Et, A1, (const float*)0, (const float*)0, (const float*)0,
          M, N, K, NP, 32, 32, 0, 0, 0, 1, 1.0f, 0);
    }
    k_kdft<<<(73728 + 255) / 256, 256, 0, stream>>>(A1, Fk, Xft);
    k_mix<<<(73728 + 255) / 256, 256, 0, stream>>>(Xft, w1r, w1i, w2r, w2i,
                                                   Oft, l);
    {
      int tot = BATCHN * WIDTH * KM * NP;
      k_invw<<<(tot + 255) / 256, 256, 0, stream>>>(Oft, Bw);
    }
    {  // pointwise conv GEMM per batch: pw[32x32] x x[32x70225] + bias -> nxt
      int M = 32, N = NP2, K = 32;
      int tn = (N + 15) / 16, tgn = (tn + 3) / 4;
      long waves = (long)(((M + 15) / 16) * tgn) * BATCHN;
      k_gemm<4><<<(int)((waves * 32 + 255) / 256), 256, 0, stream>>>(
          pw_w + (long)l * 1024, cur, nxt, (const float*)0,
          pw_b + l * 32, (const float*)0,
          M, N, K, 32, NP2, NP2, 0, (long)WIDTH * NP2, (long)WIDTH * NP2,
          BATCHN, 1.0f, 0);
    }
    {  // inverse h-DFT GEMM per (b,o): G[265x48] x Bw[48x265], += nxt, gelu
      int M = NP, N = NP, K = 2 * KM;
      int tn = (N + 15) / 16, tgn = (tn + 3) / 4;
      long waves = (long)(((M + 15) / 16) * tgn) * BATCHN * WIDTH;
      k_gemm<4><<<(int)((waves * 32 + 255) / 256), 256, 0, stream>>>(
          Gt, Bw, nxt, nxt, (const float*)0, (const float*)0,
          M, N, K, 2 * KM, NP, NP, 0, (long)2 * KM * NP, (long)NP2,
          BATCHN * WIDTH, 1.0f / (float)NP2, (l < 3) ? 1 : 0);
    }
    float* tmp = cur; cur = nxt; nxt = tmp;
  }

  // outputs: [pred_de | x_de | pred | x_en]
  float* pred = out + 2 * 98304;
  k_head<<<8192, 128, 0, stream>>>(cur, fc1_w, fc1_b, fc2_w, fc2_b, x_en, pred);
  k_finalize<<<(1572864 + 255) / 256, 256, 0, stream>>>(
      x_en, pred, out, out + 98304, out + 2 * 98304 + 1572864);
}